// LossYOLO_v2_37460704755794
// MI455X (gfx1250) — compile-verified
//
#include <hip/hip_runtime.h>
#include <math.h>

#define NUM_CLASSES 80
#define NUM_ANC     5
#define GRID_N      26
#define N_GT        8
#define HW          676       // 26*26
#define S_          81        // 1 + NUM_CLASSES
#define IMG_STRIDE  287300    // (S_+4)*NUM_ANC*HW = 425*676
#define BATCH       128
#define EPSF        1e-7f
#define REC_F       12        // floats per record
#define NREC        (N_GT*NUM_ANC)   // 40 records per image

typedef float v2f __attribute__((ext_vector_type(2)));
typedef float v8f __attribute__((ext_vector_type(8)));

__device__ __forceinline__ float sigm(float x)  { return 1.0f / (1.0f + expf(-x)); }
// stable softplus: bce(x,z) = softplus(x) - z*x
__device__ __forceinline__ float softplusf(float x) {
    return log1pf(expf(-fabsf(x))) + fmaxf(x, 0.0f);
}

// Exact 32-lane sum via V_WMMA_F32_16X16X4_F32 with B = ones, C = 0.
// A layout (16x4 f32): lane L<16 -> A[L][0],A[L][1]; lane L>=16 -> A[L-16][2],A[L-16][3].
// With a={v,0}: D[m][n] = v_m + v_{m+16} for every n.
// C/D layout: vgpr r, lanes 0-15 -> D[r][lane]; lanes 16-31 -> D[8+r][lane-16].
// Since all columns equal: sum of d[0..7] = rows 0..7 (lanes<16) or rows 8..15 (lanes>=16);
// one shfl_xor(16) gives the full 32-lane sum on every lane.
__device__ __forceinline__ float wave_sum_wmma(float v) {
    v2f a; a[0] = v;     a[1] = 0.0f;
    v2f b; b[0] = 1.0f;  b[1] = 1.0f;
    v8f c = {0.f, 0.f, 0.f, 0.f, 0.f, 0.f, 0.f, 0.f};
    v8f d = __builtin_amdgcn_wmma_f32_16x16x4_f32(false, a, false, b, (short)0, c, false, false);
    float s = ((d[0] + d[1]) + (d[2] + d[3])) + ((d[4] + d[5]) + (d[6] + d[7]));
    s += __shfl_xor(s, 16, 32);
    return s;
}

// ---------------- K1: exact sequential GT matching (one thread per image) ----
__global__ void k_match(const float* __restrict__ gboxes,
                        const int*   __restrict__ labels,
                        float* __restrict__ recs) {
    int b = blockIdx.x * blockDim.x + threadIdx.x;
    if (b >= BATCH) return;
    const float ancw[NUM_ANC] = {0.05f, 0.12f, 0.25f, 0.45f, 0.80f};
    const float anch[NUM_ANC] = {0.07f, 0.15f, 0.30f, 0.50f, 0.85f};
    const float* gb = gboxes + (size_t)b * N_GT * 4;
    const int*   lb = labels + (size_t)b * N_GT;

    int   cellid[N_GT];
    int   canon[N_GT];
    float st[N_GT][NUM_ANC][10];   // [0]=label0,[1..2]=txy,[3..4]=twh,[5]=wt,[6..9]=gbox
    bool  wr[N_GT][NUM_ANC];
    for (int s = 0; s < N_GT; s++)
        for (int a = 0; a < NUM_ANC; a++) wr[s][a] = false;

    for (int i = 0; i < N_GT; i++) {
        float l  = gb[i*4+0], t = gb[i*4+1], r = gb[i*4+2], bo = gb[i*4+3];
        float cx = (l + r) * 0.5f, cy = (t + bo) * 0.5f;
        float w  = r - l,          h  = bo - t;
        float area = w * h;
        float best = -1.0f; int besta = 0; int ign = 0;
        for (int a = 0; a < NUM_ANC; a++) {
            float inter = fminf(w, ancw[a]) * fminf(h, anch[a]);
            float iou = inter / (area + ancw[a]*anch[a] - inter + EPSF);
            if (iou > 0.5f) ign |= (1 << a);
            if (iou > best) { best = iou; besta = a; }   // first-max like jnp.argmax
        }
        int col = (int)(cx * (float)GRID_N);
        int row = (int)(cy * (float)GRID_N);
        float tx = (cx - (float)col / (float)GRID_N) * (float)GRID_N;
        float ty = (cy - (float)row / (float)GRID_N) * (float)GRID_N;
        float tw = logf(w / ancw[besta]);
        float th = logf(h / anch[besta]);
        float wt = 2.0f - area;
        cellid[i] = row * GRID_N + col;

        int s = i;
        for (int j = 0; j < i; j++)
            if (cellid[j] == cellid[i]) { s = canon[j]; break; }
        canon[i] = s;

        // reference order: first ignore-mark weight slots, then full overwrite at besta
        for (int a = 0; a < NUM_ANC; a++)
            if ((ign >> a) & 1) st[s][a][5] = -1.0f;
        wr[s][besta] = true;
        st[s][besta][0] = (float)(lb[i] - 1);
        st[s][besta][1] = tx; st[s][besta][2] = ty;
        st[s][besta][3] = tw; st[s][besta][4] = th;
        st[s][besta][5] = wt;
        st[s][besta][6] = l;  st[s][besta][7] = t;
        st[s][besta][8] = r;  st[s][besta][9] = bo;
    }

    for (int s = 0; s < N_GT; s++) {
        for (int a = 0; a < NUM_ANC; a++) {
            float* rec = recs + ((size_t)(b * N_GT + s) * NUM_ANC + a) * REC_F;
            if (wr[s][a]) {
                rec[0] = 1.0f;
                rec[1] = (float)cellid[s];
                rec[2] = st[s][a][0];
                rec[3] = st[s][a][1]; rec[4] = st[s][a][2];
                rec[5] = st[s][a][3]; rec[6] = st[s][a][4];
                rec[7] = st[s][a][5];
                rec[8] = st[s][a][6]; rec[9] = st[s][a][7];
                rec[10] = st[s][a][8]; rec[11] = st[s][a][9];
            } else {
                #pragma unroll
                for (int k = 0; k < REC_F; k++) rec[k] = 0.0f;
            }
        }
    }
}

// ---------------- K2: per-image loss terms (one block per image) -------------
__global__ __launch_bounds__(256) void k_loss(const float* __restrict__ py,
                                              const float* __restrict__ recs,
                                              float* __restrict__ part) {
    const int b = blockIdx.x;
    const float* base = py + (size_t)b * IMG_STRIDE;

    // Part A: neg-conf over ALL positions = sum sigmoid(pconf)^2.
    // pconf is the contiguous first NUM_ANC*HW floats of the image block.
    float qneg = 0.0f;
    const float4* p4 = (const float4*)base;       // 3380/4 = 845 vectors, 16B aligned
    for (int i = threadIdx.x; i < (NUM_ANC * HW) / 4; i += 256) {
        float4 v = p4[i];
        float s0 = sigm(v.x), s1 = sigm(v.y), s2 = sigm(v.z), s3 = sigm(v.w);
        qneg += s0*s0 + s1*s1 + s2*s2 + s3*s3;
    }

    // Part B: corrections + positive terms at matched records.
    float qpos = 0.f, qcls = 0.f, qtxty = 0.f, qtwth = 0.f, qnpos = 0.f;
    if (threadIdx.x < NREC) {
        const float* rec = recs + ((size_t)b * NREC + threadIdx.x) * REC_F;
        if (rec[0] > 0.5f) {
            const float ancw[NUM_ANC] = {0.05f, 0.12f, 0.25f, 0.45f, 0.80f};
            const float anch[NUM_ANC] = {0.07f, 0.15f, 0.30f, 0.50f, 0.85f};
            int a    = threadIdx.x % NUM_ANC;
            int cell = (int)rec[1];
            int lab  = (int)rec[2];
            float x0  = base[a * HW + cell];
            float xtx = base[((S_    ) * NUM_ANC + a) * HW + cell];
            float xty = base[((S_ + 1) * NUM_ANC + a) * HW + cell];
            float xtw = base[((S_ + 2) * NUM_ANC + a) * HW + cell];
            float xth = base[((S_ + 3) * NUM_ANC + a) * HW + cell];
            // decode predicted box
            float colf = (float)(cell % GRID_N), rowf = (float)(cell / GRID_N);
            float px = (sigm(xtx) + colf) / (float)GRID_N;
            float pyc = (sigm(xty) + rowf) / (float)GRID_N;
            float pw = expf(xtw) * ancw[a], ph = expf(xth) * anch[a];
            float pl = px - pw * 0.5f, pt = pyc - ph * 0.5f;
            float pr = px + pw * 0.5f, pb = pyc + ph * 0.5f;
            float gl = rec[8], gt = rec[9], gr = rec[10], gbb = rec[11];
            float iw = fmaxf(fminf(pr, gr) - fmaxf(pl, gl), 0.0f);
            float ih = fmaxf(fminf(pb, gbb) - fmaxf(pt, gt), 0.0f);
            float inter = iw * ih;
            float aa = fmaxf(pr - pl, 0.f) * fmaxf(pb - pt, 0.f);
            float ab = fmaxf(gr - gl, 0.f) * fmaxf(gbb - gt, 0.f);
            float gconf = inter / (aa + ab - inter + EPSF);
            if (gconf > 0.0f) {
                float s = sigm(x0);
                qneg  -= s * s;                       // move out of neg mask
                qpos  += (s - gconf) * (s - gconf);
                qnpos += 1.0f;
                float cl = 0.0f;
                for (int c = 0; c < NUM_CLASSES; c++) {
                    float x = base[((1 + c) * NUM_ANC + a) * HW + cell];
                    cl += softplusf(x) - ((c == lab) ? x : 0.0f);
                }
                qcls += cl;
                float wt = rec[7];
                float t1 = (softplusf(xtx) - rec[3] * xtx)
                         + (softplusf(xty) - rec[4] * xty);
                qtxty += t1 * wt;
                float d1 = xtw - rec[5], d2 = xth - rec[6];
                qtwth += (d1 * d1 + d2 * d2) * wt;
            }
        }
    }

    // Block reduction: WMMA wave-sum per quantity, then 8 waves via LDS.
    __shared__ float lsum[6][8];
    int wid = threadIdx.x >> 5;
    float t0 = wave_sum_wmma(qneg);
    float t1 = wave_sum_wmma(qpos);
    float t2 = wave_sum_wmma(qcls);
    float t3 = wave_sum_wmma(qtxty);
    float t4 = wave_sum_wmma(qtwth);
    float t5 = wave_sum_wmma(qnpos);
    if ((threadIdx.x & 31) == 0) {
        lsum[0][wid] = t0; lsum[1][wid] = t1; lsum[2][wid] = t2;
        lsum[3][wid] = t3; lsum[4][wid] = t4; lsum[5][wid] = t5;
    }
    __syncthreads();
    if (threadIdx.x < 6) {
        float s = 0.0f;
        for (int w = 0; w < 8; w++) s += lsum[threadIdx.x][w];
        part[b * 6 + threadIdx.x] = s;
    }
}

// ---------------- K3: final combine (single wave) ---------------------------
__global__ void k_final(const float* __restrict__ part, float* __restrict__ out) {
    int l = threadIdx.x;   // 32 threads
    float q[6] = {0.f, 0.f, 0.f, 0.f, 0.f, 0.f};
    for (int b = l; b < BATCH; b += 32) {
        #pragma unroll
        for (int j = 0; j < 6; j++) q[j] += part[b * 6 + j];
    }
    float tneg  = wave_sum_wmma(q[0]);
    float tpos  = wave_sum_wmma(q[1]);
    float tcls  = wave_sum_wmma(q[2]);
    float ttxty = wave_sum_wmma(q[3]);
    float ttwth = wave_sum_wmma(q[4]);
    float tnpos = wave_sum_wmma(q[5]);
    if (l == 0) {
        const float Bf = (float)BATCH;
        float loss = 5.0f * tpos / Bf
                   + 1.0f * tneg / Bf
                   + tcls / fmaxf(tnpos, 1.0f)
                   + ttxty / Bf
                   + ttwth / Bf;
        out[0] = loss;
    }
}

extern "C" void kernel_launch(void* const* d_in, const int* in_sizes, int n_in,
                              void* d_out, int out_size, void* d_ws, size_t ws_size,
                              hipStream_t stream) {
    const float* pyolos = (const float*)d_in[0];
    const float* gboxes = (const float*)d_in[1];
    const int*   labels = (const int*)d_in[2];

    float* recs = (float*)d_ws;                              // 128*40*12 floats = 240 KB
    float* part = recs + (size_t)BATCH * NREC * REC_F;       // 128*6 floats

    k_match<<<2, 64, 0, stream>>>(gboxes, labels, recs);
    k_loss<<<BATCH, 256, 0, stream>>>(pyolos, recs, part);
    k_final<<<1, 32, 0, stream>>>(part, (float*)d_out);
}